// YOWO_57166014709896
// MI455X (gfx1250) — compile-verified
//
#include <hip/hip_runtime.h>
#include <stdint.h>

#define NUM_CLASSES 80
#define MROWS 5120            // anchors per batch (32*32*5)
#define BSZ 32
#define TOPK 1000
#define NSORT 8192
#define IMG_SIZE_F 512.0f
#define STRIDE_F 16.0f
#define CONF_THRESH 0.05f
#define NMS_THRESH 0.6f

typedef __attribute__((ext_vector_type(2))) float v2f;
typedef __attribute__((ext_vector_type(8))) float v8f;

__device__ __forceinline__ float rcp_fast(float x) { return __builtin_amdgcn_rcpf(x); }
__device__ __forceinline__ float sigmoid_fast(float x) {
    return rcp_fast(1.0f + __expf(-x));      // v_exp_f32 + v_rcp_f32
}

// order-preserving float<->u32 maps (ascending u <-> ascending f)
__device__ __forceinline__ uint32_t f2u(float f) {
    uint32_t x = __float_as_uint(f);
    return (x & 0x80000000u) ? ~x : (x | 0x80000000u);
}
__device__ __forceinline__ float u2f(uint32_t u) {
    uint32_t x = (u & 0x80000000u) ? (u & 0x7FFFFFFFu) : ~u;
    return __uint_as_float(x);
}

// ---------------------------------------------------------------------------
// Kernel 1: per-row score = sigmoid(conf)/sum(exp(l - lmax)), label = argmax.
// One wave handles 16 rows. The 80-class exp-sum is reduced with
// V_WMMA_F32_16X16X4_F32 using A = all-ones: D[m][n] = sum_k B[k][n], so each
// lane's D[0] is the softmax denominator of its own row (column n = lane&15).
// Layout-robust: invariant to the exact K<->(half,VGPR) striping.
// ---------------------------------------------------------------------------
__global__ __launch_bounds__(256) void yowo_score_kernel(
    const float* __restrict__ conf, const float* __restrict__ cls,
    uint64_t* __restrict__ keys, uint32_t* __restrict__ labels)
{
    const int tid  = threadIdx.x;
    const int wave = tid >> 5;
    const int lane = tid & 31;
    const int n    = lane & 15;    // matrix column == local row id
    const int h    = lane >> 4;    // which 40-class half this lane reduces
    const uint32_t r32 = (blockIdx.x * 8u + (uint32_t)wave) * 16u + (uint32_t)n;
    const size_t r = (size_t)r32;

    const float4* p = (const float4*)(cls + r * NUM_CLASSES + (size_t)(h * 40));
    float4 v[10];
#pragma unroll
    for (int i = 0; i < 10; ++i) v[i] = p[i];

    // max + first-argmax over this lane's 40 classes (ascending index, strict >)
    float mx = v[0].x;
    int arg = h * 40;
#pragma unroll
    for (int i = 0; i < 10; ++i) {
        float a0 = v[i].x, a1 = v[i].y, a2 = v[i].z, a3 = v[i].w;
        int c = h * 40 + i * 4;
        if (a0 > mx) { mx = a0; arg = c + 0; }
        if (a1 > mx) { mx = a1; arg = c + 1; }
        if (a2 > mx) { mx = a2; arg = c + 2; }
        if (a3 > mx) { mx = a3; arg = c + 3; }
    }
    // combine the two 40-class halves (wave32 cross-half shuffle)
    float omx = __shfl_xor(mx, 16, 32);
    int  oarg = __shfl_xor(arg, 16, 32);
    if (omx > mx || (omx == mx && oarg < arg)) { mx = omx; arg = oarg; }

    // two 20-class partial exp-sums -> B matrix operand
    float b0 = 0.0f, b1 = 0.0f;
#pragma unroll
    for (int i = 0; i < 5; ++i)
        b0 += __expf(v[i].x - mx) + __expf(v[i].y - mx) +
              __expf(v[i].z - mx) + __expf(v[i].w - mx);
#pragma unroll
    for (int i = 5; i < 10; ++i)
        b1 += __expf(v[i].x - mx) + __expf(v[i].y - mx) +
              __expf(v[i].z - mx) + __expf(v[i].w - mx);

    v2f Amat; Amat[0] = 1.0f; Amat[1] = 1.0f;      // all-ones 16x4
    v2f Bmat; Bmat[0] = b0;   Bmat[1] = b1;        // partial sums, column n
    v8f Cm = {};
    v8f Dm = __builtin_amdgcn_wmma_f32_16x16x4_f32(
        false, Amat, false, Bmat, (short)0, Cm, false, false);
    float denom = Dm[0];   // column-sum of B = full 80-class exp-sum, all lanes

    if (h == 0) {
        // score = sigmoid(conf)/denom = rcp((1+exp(-conf)) * denom)
        float s = rcp_fast((1.0f + __expf(-conf[r])) * denom);
        uint32_t bidx = blockIdx.x / 40u;          // 40 blocks per batch (uniform)
        uint32_t m = r32 - bidx * (uint32_t)MROWS; // index within batch
        // ascending-u64 sort order == descending score, then ascending index
        keys[r]   = ((uint64_t)(~f2u(s)) << 32) | (uint64_t)m;
        labels[r] = (uint32_t)arg;
    }
}

// ---------------------------------------------------------------------------
// Kernel 2: per-batch fused top-k (bitonic sort in LDS) + box decode + NMS.
// One 1024-thread workgroup per batch; ~90KB dynamic LDS (CDNA5: 320KB/WGP).
// ---------------------------------------------------------------------------
extern __shared__ uint8_t smem_raw[];

__global__ __launch_bounds__(1024) void yowo_topk_nms_kernel(
    const float* __restrict__ reg, const float* __restrict__ anchors,
    const uint64_t* __restrict__ gkeys, const uint32_t* __restrict__ glabels,
    float* __restrict__ out)
{
    uint64_t* keys = (uint64_t*)smem_raw;                      // NSORT u64
    float* bx1 = (float*)(smem_raw + (size_t)NSORT * 8);       // TOPK each
    float* by1 = bx1 + TOPK;
    float* bx2 = by1 + TOPK;
    float* by2 = bx2 + TOPK;
    float* bar = by2 + TOPK;
    uint32_t* skeep = (uint32_t*)(bar + TOPK);

    const int tid = threadIdx.x;
    const int b   = blockIdx.x;
    const int BK  = BSZ * TOPK;

    for (int i = tid; i < NSORT; i += 1024)
        keys[i] = (i < MROWS) ? gkeys[(size_t)b * MROWS + i]
                              : 0xFFFFFFFFFFFFFFFFull;   // pad sorts last
    __syncthreads();

    // bitonic sort, ascending u64 keys
    for (int k = 2; k <= NSORT; k <<= 1) {
        for (int j = k >> 1; j > 0; j >>= 1) {
            for (int idx = tid; idx < NSORT; idx += 1024) {
                int ixj = idx ^ j;
                if (ixj > idx) {
                    uint64_t a = keys[idx], c = keys[ixj];
                    bool up = ((idx & k) == 0);
                    if ((a > c) == up) { keys[idx] = c; keys[ixj] = a; }
                }
            }
            __syncthreads();
        }
    }

    // gather + decode top-1000; thread tid owns box tid through NMS
    float score = 0.0f, x1 = 0.0f, y1 = 0.0f, x2 = 0.0f, y2 = 0.0f, area = 0.0f;
    if (tid < TOPK) {
        uint64_t key = keys[tid];
        uint32_t m = (uint32_t)key;
        score = u2f(~(uint32_t)(key >> 32));

        float4 rg = ((const float4*)reg)[(size_t)b * MROWS + m];
        float4 an = ((const float4*)anchors)[m];
        float cx = sigmoid_fast(rg.x) * STRIDE_F + an.x;
        float cy = sigmoid_fast(rg.y) * STRIDE_F + an.y;
        float w  = __expf(rg.z) * an.z;
        float hh = __expf(rg.w) * an.w;
        float bxl = fminf(fmaxf((cx - 0.5f * w)  * (1.0f / IMG_SIZE_F), 0.0f), 1.0f);
        float byl = fminf(fmaxf((cy - 0.5f * hh) * (1.0f / IMG_SIZE_F), 0.0f), 1.0f);
        float bxh = fminf(fmaxf((cx + 0.5f * w)  * (1.0f / IMG_SIZE_F), 0.0f), 1.0f);
        float byh = fminf(fmaxf((cy + 0.5f * hh) * (1.0f / IMG_SIZE_F), 0.0f), 1.0f);

        uint32_t lab = glabels[(size_t)b * MROWS + m];
        float off = 2.0f * (float)lab;
        x1 = bxl + off; y1 = byl + off; x2 = bxh + off; y2 = byh + off;
        area = (x2 - x1) * (y2 - y1);

        out[BK + b * TOPK + tid] = (float)lab;                 // labels
        float* bb = out + 2 * BK + ((size_t)b * TOPK + tid) * 4;
        bb[0] = bxl; bb[1] = byl; bb[2] = bxh; bb[3] = byh;    // bboxes

        bx1[tid] = x1; by1[tid] = y1; bx2[tid] = x2; by2[tid] = y2;
        bar[tid] = area; skeep[tid] = 1u;
    }
    __syncthreads();

    // sequential NMS: row i broadcast from LDS vs register-resident box tid
    for (int i = 0; i < TOPK; ++i) {
        if (skeep[i]) {                       // uniform LDS-broadcast read
            if (tid < TOPK && tid > i) {
                float xx1 = fmaxf(bx1[i], x1);
                float yy1 = fmaxf(by1[i], y1);
                float xx2 = fminf(bx2[i], x2);
                float yy2 = fminf(by2[i], y2);
                float w = fmaxf(1e-28f, xx2 - xx1);
                float h = fmaxf(1e-28f, yy2 - yy1);
                float inter = w * h;
                float iou = inter / (bar[i] + area - inter + 1e-14f);
                if (iou > NMS_THRESH) skeep[tid] = 0u;
            }
            __syncthreads();                  // only needed when writes occurred
        }
    }

    if (tid < TOPK) {
        bool kp = (skeep[tid] != 0u) && (score >= CONF_THRESH);
        out[b * TOPK + tid]          = kp ? score : 0.0f;      // scores_out
        out[6 * BK + b * TOPK + tid] = kp ? 1.0f : 0.0f;       // keep
    }
}

// ---------------------------------------------------------------------------
extern "C" void kernel_launch(void* const* d_in, const int* in_sizes, int n_in,
                              void* d_out, int out_size, void* d_ws, size_t ws_size,
                              hipStream_t stream) {
    const float* conf = (const float*)d_in[0];   // (B, M, 1)
    const float* cls  = (const float*)d_in[1];   // (B, M, 80)
    const float* reg  = (const float*)d_in[2];   // (B, M, 4)
    const float* anc  = (const float*)d_in[3];   // (M, 4)
    float* out = (float*)d_out;                  // scores|labels|bboxes|keep

    uint64_t* keys   = (uint64_t*)d_ws;                                   // B*M u64
    uint32_t* labels = (uint32_t*)((uint8_t*)d_ws + (size_t)BSZ * MROWS * 8);

    // 163840 rows / (8 waves * 16 rows) = 1280 blocks
    yowo_score_kernel<<<dim3(1280), dim3(256), 0, stream>>>(conf, cls, keys, labels);

    size_t shmem = (size_t)NSORT * 8 + (size_t)TOPK * 5 * 4 + (size_t)TOPK * 4; // 89536 B
    yowo_topk_nms_kernel<<<dim3(BSZ), dim3(1024), shmem, stream>>>(reg, anc, keys, labels, out);
}